// NystromAttention_77902116814999
// MI455X (gfx1250) — compile-verified
//
#include <hip/hip_runtime.h>
#include <hip/hip_bf16.h>

// NystromAttention for MI455X (gfx1250, wave32, WMMA).
// GEMMs: V_WMMA_F32_16X16X4_F32 with DOUBLE-BUFFERED async global->LDS staging
// (GLOBAL_LOAD_ASYNC_TO_LDS_B128, ASYNCcnt partial waits) and LDS-resident
// fragments. 64x64 block tiles, 4 waves/block, each wave 16x64 (4 accums).
// Async loads complete in-order per wave, so "s_wait_asynccnt 8" after issuing
// the next chunk's 8 copies guarantees the previous chunk's 8 have landed.

typedef __attribute__((ext_vector_type(2))) float v2f;
typedef __attribute__((ext_vector_type(8))) float v8f;

#define B_ 4
#define N_ 4096
#define D_ 512
#define H_ 8
#define DH_ 64
#define M_ 256
#define SCALE_ 0.125f  // 64^-0.5

#define KC 32          // K chunk staged in LDS
#define A_STRIDE 36    // padded row stride (floats): 144B rows -> 16B aligned, bank-spread
#define B_STRIDE 68    // padded row stride for NN B tiles (272B rows, 16B aligned)

// -------------------- CDNA5 async copy (global -> LDS) -----------------------

__device__ __forceinline__ void async_copy_b128(unsigned lds_off, const float* src) {
  // per-lane: copies 16 bytes from the global address src to LDS offset lds_off; tracked by ASYNCcnt
  asm volatile("global_load_async_to_lds_b128 %0, %1, off"
               :
               : "v"(lds_off), "v"(src)
               : "memory");
}

template <int CNT>
__device__ __forceinline__ void wait_async_le() {
  asm volatile("s_wait_asynccnt %0" ::"i"(CNT) : "memory");
}

__device__ __forceinline__ unsigned lds_offset(const void* p) {
  // the low 32 bits of a generic LDS pointer are the LDS offset
  return (unsigned)(size_t)p;
}

// ------------------------ staging + compute helpers --------------------------

// A tile: 64 rows x KC (row stride A_STRIDE); B tile (NN): KC x 64 (B_STRIDE).
__device__ __forceinline__ void stage_nn(const float* __restrict__ A, int lda,
                                         const float* __restrict__ B, int ldb,
                                         int mblk, int nblk, int kc, int tid,
                                         unsigned ash, unsigned bsh) {
#pragma unroll
  for (int rnd = 0; rnd < 4; ++rnd) {
    const int idx = rnd * 512 + tid * 4;
    const int ar = idx >> 5, ac = idx & 31;
    async_copy_b128(ash + (unsigned)(ar * A_STRIDE + ac) * 4u,
                    A + (size_t)(mblk + ar) * lda + kc + ac);
    const int bk = idx >> 6, bn = idx & 63;
    async_copy_b128(bsh + (unsigned)(bk * B_STRIDE + bn) * 4u,
                    B + (size_t)(kc + bk) * ldb + nblk + bn);
  }
}

// NT: both tiles are [64][KC] row-major with A_STRIDE padding.
__device__ __forceinline__ void stage_nt(const float* __restrict__ A, int lda,
                                         const float* __restrict__ Bt, int ldbt,
                                         int mblk, int nblk, int kc, int tid,
                                         unsigned ash, unsigned bsh) {
#pragma unroll
  for (int rnd = 0; rnd < 4; ++rnd) {
    const int idx = rnd * 512 + tid * 4;
    const int r = idx >> 5, c = idx & 31;
    async_copy_b128(ash + (unsigned)(r * A_STRIDE + c) * 4u,
                    A + (size_t)(mblk + r) * lda + kc + c);
    async_copy_b128(bsh + (unsigned)(r * A_STRIDE + c) * 4u,
                    Bt + (size_t)(nblk + r) * ldbt + kc + c);
  }
}

__device__ __forceinline__ void compute_nn(const float* __restrict__ Ash,
                                           const float* __restrict__ Bsh,
                                           int wave, int half, int l16, v8f acc[4]) {
#pragma unroll
  for (int k = 0; k < KC; k += 4) {
    const int ka = k + 2 * half;
    const v2f a = *(const v2f*)&Ash[(wave * 16 + l16) * A_STRIDE + ka];
#pragma unroll
    for (int t = 0; t < 4; ++t) {
      v2f b;
      b[0] = Bsh[ka * B_STRIDE + t * 16 + l16];
      b[1] = Bsh[(ka + 1) * B_STRIDE + t * 16 + l16];
      acc[t] = __builtin_amdgcn_wmma_f32_16x16x4_f32(false, a, false, b, (short)0,
                                                     acc[t], false, false);
    }
  }
}

__device__ __forceinline__ void compute_nt(const float* __restrict__ Ash,
                                           const float* __restrict__ Bsh,
                                           int wave, int half, int l16, v8f acc[4]) {
#pragma unroll
  for (int k = 0; k < KC; k += 4) {
    const int ka = k + 2 * half;
    const v2f a = *(const v2f*)&Ash[(wave * 16 + l16) * A_STRIDE + ka];
#pragma unroll
    for (int t = 0; t < 4; ++t) {
      const v2f b = *(const v2f*)&Bsh[(t * 16 + l16) * A_STRIDE + ka];
      acc[t] = __builtin_amdgcn_wmma_f32_16x16x4_f32(false, a, false, b, (short)0,
                                                     acc[t], false, false);
    }
  }
}

// --------------------------- GEMM kernels ------------------------------------
// block = 128 threads (4 waves); block tile 64M x 64N; wave w owns rows
// [mblk+16w, +16) across all 64 columns. Double-buffered LDS pipeline.

__global__ void gemm_nn_kernel(const float* __restrict__ A, const float* __restrict__ B,
                               float* __restrict__ C, int K, int lda, int ldb, int ldc,
                               long long sA, long long sB, long long sC) {
  __shared__ float Ash[2][64 * A_STRIDE];
  __shared__ float Bsh[2][KC * B_STRIDE];
  const long long bh = blockIdx.z;
  A += bh * sA; B += bh * sB; C += bh * sC;
  const int tid = threadIdx.x;
  const int lane = tid & 31, wave = tid >> 5;
  const int half = lane >> 4, l16 = lane & 15;
  const int mblk = blockIdx.y * 64;
  const int nblk = blockIdx.x * 64;
  const unsigned ash[2] = {lds_offset(Ash[0]), lds_offset(Ash[1])};
  const unsigned bsh[2] = {lds_offset(Bsh[0]), lds_offset(Bsh[1])};

  v8f acc[4] = {{}, {}, {}, {}};
  const int nch = K / KC;
  stage_nn(A, lda, B, ldb, mblk, nblk, 0, tid, ash[0], bsh[0]);
  for (int i = 0; i < nch; ++i) {
    if (i + 1 < nch) {
      stage_nn(A, lda, B, ldb, mblk, nblk, (i + 1) * KC, tid, ash[(i + 1) & 1], bsh[(i + 1) & 1]);
      wait_async_le<8>();   // chunk i's 8 copies done; chunk i+1's 8 in flight
    } else {
      wait_async_le<0>();
    }
    __syncthreads();
    compute_nn(Ash[i & 1], Bsh[i & 1], wave, half, l16, acc);
    __syncthreads();        // buffer (i&1) free for chunk i+2's copies
  }
  const int m0 = mblk + wave * 16;
#pragma unroll
  for (int t = 0; t < 4; ++t)
#pragma unroll
    for (int i = 0; i < 8; ++i)
      C[(size_t)(m0 + i + 8 * half) * ldc + nblk + t * 16 + l16] = acc[t][i];
}

__global__ void gemm_nt_kernel(const float* __restrict__ A, const float* __restrict__ Bt,
                               float* __restrict__ C, int K, int lda, int ldbt, int ldc,
                               long long sA, long long sB, long long sC) {
  __shared__ float Ash[2][64 * A_STRIDE];
  __shared__ float Bsh[2][64 * A_STRIDE];
  const long long bh = blockIdx.z;
  A += bh * sA; Bt += bh * sB; C += bh * sC;
  const int tid = threadIdx.x;
  const int lane = tid & 31, wave = tid >> 5;
  const int half = lane >> 4, l16 = lane & 15;
  const int mblk = blockIdx.y * 64;
  const int nblk = blockIdx.x * 64;
  const unsigned ash[2] = {lds_offset(Ash[0]), lds_offset(Ash[1])};
  const unsigned bsh[2] = {lds_offset(Bsh[0]), lds_offset(Bsh[1])};

  v8f acc[4] = {{}, {}, {}, {}};
  const int nch = K / KC;
  stage_nt(A, lda, Bt, ldbt, mblk, nblk, 0, tid, ash[0], bsh[0]);
  for (int i = 0; i < nch; ++i) {
    if (i + 1 < nch) {
      stage_nt(A, lda, Bt, ldbt, mblk, nblk, (i + 1) * KC, tid, ash[(i + 1) & 1], bsh[(i + 1) & 1]);
      wait_async_le<8>();
    } else {
      wait_async_le<0>();
    }
    __syncthreads();
    compute_nt(Ash[i & 1], Bsh[i & 1], wave, half, l16, acc);
    __syncthreads();
  }
  const int m0 = mblk + wave * 16;
#pragma unroll
  for (int t = 0; t < 4; ++t)
#pragma unroll
    for (int i = 0; i < 8; ++i)
      C[(size_t)(m0 + i + 8 * half) * ldc + nblk + t * 16 + l16] = acc[t][i];
}

// qkv GEMM: x[16384,512] @ Wqkv[512,1536]; epilogue splits into q(scaled)/k/v
// in [b,h,n,dh] layout. Each wave's 16-col sub-tile never crosses a 64 (head)
// or 512 (q/k/v) boundary, so the epilogue is wave-uniform.
__global__ void gemm_qkv_kernel(const float* __restrict__ x, const float* __restrict__ W,
                                float* __restrict__ q, float* __restrict__ k,
                                float* __restrict__ v) {
  __shared__ float Ash[2][64 * A_STRIDE];
  __shared__ float Bsh[2][KC * B_STRIDE];
  const int tid = threadIdx.x;
  const int lane = tid & 31, wave = tid >> 5;
  const int half = lane >> 4, l16 = lane & 15;
  const int mblk = blockIdx.y * 64;
  const int nblk = blockIdx.x * 64;
  const unsigned ash[2] = {lds_offset(Ash[0]), lds_offset(Ash[1])};
  const unsigned bsh[2] = {lds_offset(Bsh[0]), lds_offset(Bsh[1])};

  v8f acc[4] = {{}, {}, {}, {}};
  const int nch = D_ / KC;  // 16
  stage_nn(x, D_, W, 3 * H_ * DH_, mblk, nblk, 0, tid, ash[0], bsh[0]);
  for (int i = 0; i < nch; ++i) {
    if (i + 1 < nch) {
      stage_nn(x, D_, W, 3 * H_ * DH_, mblk, nblk, (i + 1) * KC, tid,
               ash[(i + 1) & 1], bsh[(i + 1) & 1]);
      wait_async_le<8>();
    } else {
      wait_async_le<0>();
    }
    __syncthreads();
    compute_nn(Ash[i & 1], Bsh[i & 1], wave, half, l16, acc);
    __syncthreads();
  }
  const int m0 = mblk + wave * 16;
#pragma unroll
  for (int t = 0; t < 4; ++t) {
    const int col = nblk + t * 16 + l16;
    const int which = col >> 9;           // 0=q 1=k 2=v
    const int hd = col & 511;
    const int head = hd >> 6;
    const int dcol = hd & 63;
    const float scale = (which == 0) ? SCALE_ : 1.0f;
    float* dst = (which == 0) ? q : (which == 1) ? k : v;
#pragma unroll
    for (int i = 0; i < 8; ++i) {
      const int row = m0 + i + 8 * half;
      const int batch = row >> 12, tok = row & (N_ - 1);
      dst[(((size_t)batch * H_ + head) * N_ + tok) * DH_ + dcol] = acc[t][i] * scale;
    }
  }
}

// ------------------------- elementwise kernels -------------------------------

__global__ void landmark_mean_kernel(const float* __restrict__ q, const float* __restrict__ k,
                                     float* __restrict__ qland, float* __restrict__ kland) {
  const size_t total = (size_t)B_ * H_ * M_ * DH_;
  size_t idx = (size_t)blockIdx.x * blockDim.x + threadIdx.x;
  if (idx >= total) return;
  const int dcol = idx & 63;
  const size_t r = idx >> 6;
  const int lm = r & (M_ - 1);
  const size_t bh = r >> 8;
  const size_t base = (bh * N_ + (size_t)lm * 16) * DH_ + dcol;
  float sq = 0.f, sk = 0.f;
#pragma unroll
  for (int t = 0; t < 16; ++t) { sq += q[base + (size_t)t * DH_]; sk += k[base + (size_t)t * DH_]; }
  qland[idx] = sq * (1.f / 16.f);
  kland[idx] = sk * (1.f / 16.f);
}

__global__ void softmax_rows_kernel(float* __restrict__ data, int cols) {
  __shared__ float red[256];
  const size_t row = blockIdx.x;
  float* p = data + row * (size_t)cols;
  const int tid = threadIdx.x;
  float m = -3.4e38f;
  for (int c = tid; c < cols; c += 256) m = fmaxf(m, p[c]);
  red[tid] = m; __syncthreads();
  for (int s = 128; s > 0; s >>= 1) {
    if (tid < s) red[tid] = fmaxf(red[tid], red[tid + s]);
    __syncthreads();
  }
  m = red[0]; __syncthreads();
  float sum = 0.f;
  for (int c = tid; c < cols; c += 256) { float e = __expf(p[c] - m); p[c] = e; sum += e; }
  red[tid] = sum; __syncthreads();
  for (int s = 128; s > 0; s >>= 1) {
    if (tid < s) red[tid] += red[tid + s];
    __syncthreads();
  }
  const float inv = 1.f / red[0];
  for (int c = tid; c < cols; c += 256) p[c] *= inv;
}

__global__ void init_scal_kernel(float* scal) { scal[0] = 0.f; scal[1] = 0.f; }

// max over all (b,h) of row-sums and of col-sums of attn2 (entries >= 0, so
// int-compare atomicMax on the bit pattern is order-preserving).
__global__ void pinv_scale_reduce_kernel(const float* __restrict__ a, float* __restrict__ scal) {
  const int bh = blockIdx.x, i = threadIdx.x;
  const float* p = a + (size_t)bh * M_ * M_;
  float rs = 0.f, cs = 0.f;
  for (int j = 0; j < M_; ++j) { rs += p[(size_t)i * M_ + j]; cs += p[(size_t)j * M_ + i]; }
  __shared__ float rr[256], cc[256];
  rr[i] = rs; cc[i] = cs; __syncthreads();
  for (int s = 128; s > 0; s >>= 1) {
    if (i < s) { rr[i] = fmaxf(rr[i], rr[i + s]); cc[i] = fmaxf(cc[i], cc[i + s]); }
    __syncthreads();
  }
  if (i == 0) {
    atomicMax((int*)&scal[0], __float_as_int(rr[0]));
    atomicMax((int*)&scal[1], __float_as_int(cc[0]));
  }
}

__global__ void pinv_init_kernel(const float* __restrict__ a, const float* __restrict__ scal,
                                 float* __restrict__ z) {
  const size_t total = (size_t)B_ * H_ * M_ * M_;
  size_t idx = (size_t)blockIdx.x * blockDim.x + threadIdx.x;
  if (idx >= total) return;
  const size_t bh = idx >> 16;
  const int r = (idx >> 8) & 255, c = idx & 255;
  const float inv = 1.0f / (scal[0] * scal[1]);
  z[idx] = a[(bh << 16) + (size_t)c * M_ + r] * inv;
}

__global__ void axpby_kernel(const float* __restrict__ A, const float* __restrict__ Bm,
                             float* __restrict__ C, float alpha, float beta, size_t n) {
  size_t idx = (size_t)blockIdx.x * blockDim.x + threadIdx.x;
  if (idx >= n) return;
  C[idx] = alpha * A[idx] + beta * Bm[idx];
}

// out_perm[b,t,h*64+d] = attnout[b,h,t,d] + depthwise_conv33(v)[b,h,t,d]
__global__ void residual_permute_kernel(const float* __restrict__ attnout,
                                        const float* __restrict__ v,
                                        const float* __restrict__ Wres,
                                        float* __restrict__ outp) {
  const size_t total = (size_t)B_ * H_ * N_ * DH_;
  size_t idx = (size_t)blockIdx.x * blockDim.x + threadIdx.x;
  if (idx >= total) return;
  const int d = idx & 63;
  const size_t r = idx >> 6;
  const int t = r & (N_ - 1);
  const size_t bh = r >> 12;
  const int h = bh & (H_ - 1);
  const size_t bb = bh >> 3;
  float acc = attnout[idx];
  const float* wr = Wres + h * 33;
  const float* vp = v + bh * N_ * DH_ + d;
#pragma unroll
  for (int kk = 0; kk < 33; ++kk) {
    const int tt = t + kk - 16;
    if (tt >= 0 && tt < N_) acc += wr[kk] * vp[(size_t)tt * DH_];
  }
  outp[((bb * N_ + t) * (size_t)(H_ * DH_)) + h * DH_ + d] = acc;
}

// ------------------------------ host side ------------------------------------

extern "C" void kernel_launch(void* const* d_in, const int* in_sizes, int n_in,
                              void* d_out, int out_size, void* d_ws, size_t ws_size,
                              hipStream_t stream) {
  (void)in_sizes; (void)n_in; (void)out_size; (void)ws_size;
  const float* x    = (const float*)d_in[0];
  const float* Wqkv = (const float*)d_in[1];
  const float* Wout = (const float*)d_in[2];
  const float* Wres = (const float*)d_in[3];
  float* out = (float*)d_out;

  float* ws = (float*)d_ws;
  size_t off = 0;
  const size_t szHND = (size_t)B_ * H_ * N_ * DH_;    // 8388608
  const size_t szHMD = (size_t)B_ * H_ * M_ * DH_;    // 524288
  const size_t szHNM = (size_t)B_ * H_ * N_ * M_;     // 33554432
  const size_t szHMM = (size_t)B_ * H_ * M_ * M_;     // 2097152
  float* q      = ws + off; off += szHND;
  float* k      = ws + off; off += szHND;
  float* v      = ws + off; off += szHND;
  float* qland  = ws + off; off += szHMD;
  float* kland  = ws + off; off += szHMD;
  float* attn1  = ws + off; off += szHNM;
  float* attn3  = ws + off; off += szHNM;
  float* attn2  = ws + off; off += szHMM;
  float* z0     = ws + off; off += szHMM;
  float* z1     = ws + off; off += szHMM;
  float* px     = ws + off; off += szHMM;
  float* pt1    = ws + off; off += szHMM;
  float* pt2    = ws + off; off += szHMM;
  float* a3v    = ws + off; off += szHMD;
  float* zv     = ws + off; off += szHMD;
  float* attnout= ws + off; off += szHND;
  float* outperm= ws + off; off += szHND;
  float* scal   = ws + off; off += 2;

  const long long sMM = (long long)M_ * M_;
  const long long sMD = (long long)M_ * DH_;
  const long long sND = (long long)N_ * DH_;
  const long long sNM = (long long)N_ * M_;

  // 1) qkv projection + head split + q scaling  (M=16384, N=1536, K=512)
  gemm_qkv_kernel<<<dim3(24, 256, 1), 128, 0, stream>>>(x, Wqkv, q, k, v);
  // 2) landmark means
  landmark_mean_kernel<<<(unsigned)(szHMD / 256), 256, 0, stream>>>(q, k, qland, kland);
  // 3) sim2 = q_land @ k_land^T; softmax   (M=256, N=256, K=64)
  gemm_nt_kernel<<<dim3(4, 4, 32), 128, 0, stream>>>(qland, kland, attn2,
      DH_, DH_, DH_, M_, sMD, sMD, sMM);
  softmax_rows_kernel<<<32 * M_, 256, 0, stream>>>(attn2, M_);
  // 4) sim1 = q @ k_land^T; softmax        (M=4096, N=256, K=64)
  gemm_nt_kernel<<<dim3(4, 64, 32), 128, 0, stream>>>(q, kland, attn1,
      DH_, DH_, DH_, M_, sND, sMD, sNM);
  softmax_rows_kernel<<<32 * N_, 256, 0, stream>>>(attn1, M_);
  // 5) sim3 = q_land @ k^T; softmax        (M=256, N=4096, K=64)
  gemm_nt_kernel<<<dim3(64, 4, 32), 128, 0, stream>>>(qland, k, attn3,
      DH_, DH_, DH_, N_, sMD, sND, (long long)M_ * N_);
  softmax_rows_kernel<<<32 * M_, 256, 0, stream>>>(attn3, N_);
  // 6) Moore-Penrose pinv of attn2 (Newton-Schulz, 6 iters)
  init_scal_kernel<<<1, 1, 0, stream>>>(scal);
  pinv_scale_reduce_kernel<<<32, 256, 0, stream>>>(attn2, scal);
  pinv_init_kernel<<<(unsigned)(szHMM / 256), 256, 0, stream>>>(attn2, scal, z0);
  float* zin = z0; float* zout = z1;
  for (int it = 0; it < 6; ++it) {
    // X = A @ Z
    gemm_nn_kernel<<<dim3(4, 4, 32), 128, 0, stream>>>(attn2, zin, px,
        M_, M_, M_, M_, sMM, sMM, sMM);
    // XX = X @ X ; A2 = 7X - XX
    gemm_nn_kernel<<<dim3(4, 4, 32), 128, 0, stream>>>(px, px, pt1,
        M_, M_, M_, M_, sMM, sMM, sMM);
    axpby_kernel<<<(unsigned)(szHMM / 256), 256, 0, stream>>>(px, pt1, pt1, 7.f, -1.f, szHMM);
    // A4 = 15X - X @ A2
    gemm_nn_kernel<<<dim3(4, 4, 32), 128, 0, stream>>>(px, pt1, pt2,
        M_, M_, M_, M_, sMM, sMM, sMM);
    axpby_kernel<<<(unsigned)(szHMM / 256), 256, 0, stream>>>(px, pt2, pt2, 15.f, -1.f, szHMM);
    // Z' = 0.25 * Z @ (13I - A4) = 3.25 Z - 0.25 (Z @ A4)
    gemm_nn_kernel<<<dim3(4, 4, 32), 128, 0, stream>>>(zin, pt2, pt1,
        M_, M_, M_, M_, sMM, sMM, sMM);
    axpby_kernel<<<(unsigned)(szHMM / 256), 256, 0, stream>>>(zin, pt1, zout, 3.25f, -0.25f, szHMM);
    float* tswap = zin; zin = zout; zout = tswap;
  }
  // 7) attn3 @ v -> [m,dh]                 (M=256, N=64, K=4096)
  gemm_nn_kernel<<<dim3(1, 4, 32), 128, 0, stream>>>(attn3, v, a3v,
      N_, N_, DH_, DH_, (long long)M_ * N_, sND, sMD);
  // 8) z @ (attn3 v)                       (M=256, N=64, K=256)
  gemm_nn_kernel<<<dim3(1, 4, 32), 128, 0, stream>>>(zin, a3v, zv,
      M_, M_, DH_, DH_, sMM, sMD, sMD);
  // 9) attn1 @ zv -> [n,dh]                (M=4096, N=64, K=256)
  gemm_nn_kernel<<<dim3(1, 64, 32), 128, 0, stream>>>(attn1, zv, attnout,
      M_, M_, DH_, DH_, sNM, sMD, sND);
  // 10) + depthwise residual conv, merge heads to [b,n,512]
  residual_permute_kernel<<<(unsigned)(szHND / 256), 256, 0, stream>>>(attnout, v, Wres, outperm);
  // 11) output projection [16384,512] @ [512,512] -> d_out
  gemm_nn_kernel<<<dim3(8, 256, 1), 128, 0, stream>>>(outperm, Wout, out,
      D_, D_, D_, D_, 0, 0, 0);
}